// TopKTopPSampler_8383776161950
// MI455X (gfx1250) — compile-verified
//
#include <hip/hip_runtime.h>
#include <hip/hip_bf16.h>

#define NTHREADS 1024
#define NBINS    4096
#define CAND_MAX 512
#define CHUNK    (NTHREADS * 4)   // floats per staged chunk (16 KB)

#if defined(__AMDGCN__) && __has_builtin(__builtin_amdgcn_global_load_async_to_lds_b128)
#define HAVE_ASYNC 1
#else
#define HAVE_ASYNC 0
#endif

// Order-preserving float -> uint key (ascending key == ascending float)
__device__ __forceinline__ unsigned ordKey(float f) {
  unsigned u = __float_as_uint(f);
  return (u & 0x80000000u) ? ~u : (u | 0x80000000u);
}
__device__ __forceinline__ float keyToFloat(unsigned k) {
  unsigned u = (k & 0x80000000u) ? (k ^ 0x80000000u) : ~k;
  return __uint_as_float(u);
}

#if HAVE_ASYNC
typedef int v4i __attribute__((ext_vector_type(4)));
typedef __attribute__((address_space(1))) void as1_void;
typedef __attribute__((address_space(3))) void as3_void;
typedef __attribute__((address_space(1))) v4i as1_v4i;
typedef __attribute__((address_space(3))) v4i as3_v4i;

__device__ __forceinline__ void async_ld16(const float* g, float4* l) {
  // gfx1250: GLOBAL_LOAD_ASYNC_TO_LDS_B128 (per-lane 16B, tracked by ASYNCcnt)
  as1_void* gp = (as1_void*)(g);
  as3_void* lp = (as3_void*)(l);
  __builtin_amdgcn_global_load_async_to_lds_b128((as1_v4i*)gp, (as3_v4i*)lp, 0, 0);
}
template <int N> __device__ __forceinline__ void wait_async() {
#if __has_builtin(__builtin_amdgcn_s_wait_asynccnt)
  __builtin_amdgcn_s_wait_asynccnt(N);
#else
  asm volatile("s_wait_asynccnt %0" :: "n"(N) : "memory");
#endif
  asm volatile("" ::: "memory");
}
#endif

__global__ __launch_bounds__(NTHREADS, 1) void topk_topp_sample_kernel(
    const float* __restrict__ logits, const int* __restrict__ k_arr,
    const float* __restrict__ p_arr, const float* __restrict__ noise,
    float* __restrict__ out, int V) {
  const int row = blockIdx.x;
  const int tid = threadIdx.x;
  const float* rowp   = logits + (size_t)row * (size_t)V;
  const float* noisep = noise  + (size_t)row * (size_t)V;

  __shared__ unsigned hist[NBINS];                         // 16 KB
#if HAVE_ASYNC
  __shared__ float4 stage[2][NTHREADS];                    // 32 KB
#endif
  __shared__ unsigned uScan[NTHREADS];                     // 4 KB (reused as float scan)
  __shared__ unsigned long long candPacked[CAND_MAX];      // 4 KB
  __shared__ unsigned long long sortedPacked[CAND_MAX];    // 4 KB
  __shared__ float scoreArr[CAND_MAX];                     // 2 KB
  __shared__ int   idxArr[CAND_MAX];                       // 2 KB
  __shared__ unsigned s_cnt, s_bstar, s_mtotal;

  for (int i = tid; i < NBINS; i += NTHREADS) hist[i] = 0u;
  if (tid == 0) s_cnt = 0u;
  __syncthreads();

  const bool v4ok = ((V & 3) == 0) && ((((size_t)rowp) & 15) == 0);
  const int nFull = v4ok ? (V / CHUNK) : 0;

  // ---------- Pass 1: histogram of top 12 key bits ----------
#if HAVE_ASYNC
  if (nFull > 0) async_ld16(rowp + (size_t)tid * 4, &stage[0][tid]);
  for (int c = 0; c < nFull; ++c) {
    const int nxt = c + 1;
    if (nxt < nFull) {
      async_ld16(rowp + (size_t)nxt * CHUNK + (size_t)tid * 4, &stage[nxt & 1][tid]);
      wait_async<1>();
    } else {
      wait_async<0>();
    }
    float4 v = stage[c & 1][tid];
    atomicAdd(&hist[ordKey(v.x) >> 20], 1u);
    atomicAdd(&hist[ordKey(v.y) >> 20], 1u);
    atomicAdd(&hist[ordKey(v.z) >> 20], 1u);
    atomicAdd(&hist[ordKey(v.w) >> 20], 1u);
  }
#else
  {
    const float4* rowv = (const float4*)rowp;
    for (int i = tid; i < nFull * NTHREADS; i += NTHREADS) {
      float4 v = rowv[i];
      atomicAdd(&hist[ordKey(v.x) >> 20], 1u);
      atomicAdd(&hist[ordKey(v.y) >> 20], 1u);
      atomicAdd(&hist[ordKey(v.z) >> 20], 1u);
      atomicAdd(&hist[ordKey(v.w) >> 20], 1u);
    }
  }
#endif
  for (int i = nFull * CHUNK + tid; i < V; i += NTHREADS)
    atomicAdd(&hist[ordKey(rowp[i]) >> 20], 1u);
  __syncthreads();

  int kk = k_arr[row];
  if (kk < 1) kk = 1;
  if (kk > V) kk = V;
  const unsigned uk = (unsigned)kk;

  // ---------- suffix scan: find bin of the k-th largest ----------
  {
    const int b4 = tid * 4;
    uScan[tid] = hist[b4] + hist[b4 + 1] + hist[b4 + 2] + hist[b4 + 3];
    for (int off = 1; off < NTHREADS; off <<= 1) {
      __syncthreads();
      unsigned add = (tid + off < NTHREADS) ? uScan[tid + off] : 0u;
      __syncthreads();
      uScan[tid] += add;
    }
    __syncthreads();
    unsigned mySuf = uScan[tid];
    unsigned nxtSuf = (tid + 1 < NTHREADS) ? uScan[tid + 1] : 0u;
    if (mySuf >= uk && nxtSuf < uk) {   // unique crossing chunk
      unsigned cum = nxtSuf;
      for (int i = 3; i >= 0; --i) {
        cum += hist[b4 + i];
        if (cum >= uk) { s_bstar = (unsigned)(b4 + i); s_mtotal = cum; break; }
      }
    }
  }
  __syncthreads();
  const unsigned bstar = s_bstar;

  // ---------- Pass 2 (L2-hot): gather candidate superset ----------
  {
    const float4* rowv = (const float4*)rowp;
    const int nvec = v4ok ? (V >> 2) : 0;
    for (int i = tid; i < nvec; i += NTHREADS) {
      if (i + NTHREADS < nvec) __builtin_prefetch(&rowv[i + NTHREADS], 0, 0);
      float4 v = rowv[i];
      unsigned kx = ordKey(v.x), ky = ordKey(v.y), kz = ordKey(v.z), kw = ordKey(v.w);
      if ((kx >> 20) >= bstar) { unsigned q = atomicAdd(&s_cnt, 1u); if (q < CAND_MAX) candPacked[q] = ((unsigned long long)kx << 32) | (unsigned)(i * 4 + 0); }
      if ((ky >> 20) >= bstar) { unsigned q = atomicAdd(&s_cnt, 1u); if (q < CAND_MAX) candPacked[q] = ((unsigned long long)ky << 32) | (unsigned)(i * 4 + 1); }
      if ((kz >> 20) >= bstar) { unsigned q = atomicAdd(&s_cnt, 1u); if (q < CAND_MAX) candPacked[q] = ((unsigned long long)kz << 32) | (unsigned)(i * 4 + 2); }
      if ((kw >> 20) >= bstar) { unsigned q = atomicAdd(&s_cnt, 1u); if (q < CAND_MAX) candPacked[q] = ((unsigned long long)kw << 32) | (unsigned)(i * 4 + 3); }
    }
    for (int i = (nvec << 2) + tid; i < V; i += NTHREADS) {
      unsigned kx = ordKey(rowp[i]);
      if ((kx >> 20) >= bstar) { unsigned q = atomicAdd(&s_cnt, 1u); if (q < CAND_MAX) candPacked[q] = ((unsigned long long)kx << 32) | (unsigned)i; }
    }
  }
  __syncthreads();
  unsigned cnt = s_cnt;
  const int m = (cnt < (unsigned)CAND_MAX) ? (int)cnt : CAND_MAX;

  // ---------- rank-sort candidates by (value, index) ascending ----------
  {
    unsigned long long my = 0ull;
    int rank = 0;
    if (tid < m) my = candPacked[tid];
    for (int i = 0; i < m; ++i) {                  // broadcast LDS reads
      unsigned long long o = candPacked[i];
      if (tid < m && o < my) ++rank;
    }
    __syncthreads();
    if (tid < m) sortedPacked[rank] = my;
  }
  __syncthreads();

  const int keff = (kk < m) ? kk : m;
  const unsigned thrKey = (unsigned)(sortedPacked[m - keff] >> 32);  // exact k-th largest
  const float M = keyToFloat((unsigned)(sortedPacked[m - 1] >> 32)); // row max
  const float p = p_arr[row];

  float e = 0.f, vJ = 0.f;
  int idxJ = 0;
  bool kept1 = false;
  if (tid < m) {
    unsigned long long sp = sortedPacked[tid];
    unsigned keyJ = (unsigned)(sp >> 32);
    idxJ = (int)(sp & 0xffffffffull);
    vJ = keyToFloat(keyJ);
    kept1 = (keyJ >= thrKey);          // value-based keep => reference tie semantics
    e = kept1 ? expf(vJ - M) : 0.f;
  }

  // ---------- inclusive prefix sum of e (ascending order) ----------
  float* fScan = (float*)uScan;
  if (tid < CAND_MAX) fScan[tid] = e;
  __syncthreads();
  for (int off = 1; off < CAND_MAX; off <<= 1) {
    float add = 0.f;
    if (tid < CAND_MAX && tid >= off) add = fScan[tid - off];
    __syncthreads();
    if (tid < CAND_MAX) fScan[tid] += add;
    __syncthreads();
  }
  const float S = fScan[m - 1];
  const float cj = (tid < m) ? fScan[tid] : 0.f;

  // top-p: drop low tail with cumsum <= (1-p); always keep the max token
  bool kept2 = false;
  if (tid < m && kept1) kept2 = (cj > (1.0f - p) * S) || (tid == m - 1);

  float sc = -1.0f;
  if (kept2) sc = e / noisep[idxJ];    // argmax-equivalent to probs/exp_noise
  if (tid < CAND_MAX) {
    scoreArr[tid] = (tid < m) ? sc : -1.0f;
    idxArr[tid]   = (tid < m) ? idxJ : 0x7fffffff;
  }
  __syncthreads();
  for (int s = CAND_MAX / 2; s > 0; s >>= 1) {
    if (tid < s) {
      float a = scoreArr[tid], b = scoreArr[tid + s];
      int ia = idxArr[tid], ib = idxArr[tid + s];
      if (b > a || (b == a && ib < ia)) { scoreArr[tid] = b; idxArr[tid] = ib; }
    }
    __syncthreads();
  }
  if (tid == 0) out[row] = (float)idxArr[0];
}

extern "C" void kernel_launch(void* const* d_in, const int* in_sizes, int n_in,
                              void* d_out, int out_size, void* d_ws, size_t ws_size,
                              hipStream_t stream) {
  const float* logits = (const float*)d_in[0];
  const int*   k_arr  = (const int*)d_in[1];
  const float* p_arr  = (const float*)d_in[2];
  const float* noise  = (const float*)d_in[3];
  float* out = (float*)d_out;
  const int B = in_sizes[1];          // k has one entry per row
  const int V = in_sizes[0] / B;
  topk_topp_sample_kernel<<<B, NTHREADS, 0, stream>>>(logits, k_arr, p_arr, noise, out, V);
}